// RectangleGNN_12979391169444
// MI455X (gfx1250) — compile-verified
//
#include <hip/hip_runtime.h>

// GCN (3 layers) for MI455X / gfx1250.
// Dense node transforms use V_WMMA_F32_16X16X4_F32 (exact f32 math, wave32),
// with bias + self-loop init fused into the WMMA store path.
// Edge aggregation uses hardware fp32 global atomics (the bandwidth-bound part).

typedef float v2f __attribute__((ext_vector_type(2)));
typedef float v8f __attribute__((ext_vector_type(8)));

#define TPB 256

// ---------------- degree / norm ----------------

__global__ void init_deg(float* __restrict__ deg, int n) {
  int i = blockIdx.x * TPB + threadIdx.x;
  if (i < n) deg[i] = 1.0f;  // self-loop contributes 1 to every node's degree
}

__global__ void accum_deg(const int* __restrict__ dst, float* __restrict__ deg, int e) {
  int i = blockIdx.x * TPB + threadIdx.x;
  if (i < e) unsafeAtomicAdd(&deg[dst[i]], 1.0f);
}

__global__ void finish_dinv(float* __restrict__ deg, int n) {
  int i = blockIdx.x * TPB + threadIdx.x;
  if (i < n) {
    float d = deg[i];
    deg[i] = (d > 0.0f) ? rsqrtf(d) : 0.0f;
  }
}

// ---- dense transform: H = (relu?)A @ W via f32 WMMA, fused self-loop init ----
// One wave computes one 16-row output tile (all F columns, F<=32).
// A: nRows x Kin row-major.  W: Kin x F row-major.
// Writes H[row][n] = h  and  OutInit[row][n] = bias[n] + dinv[row]^2 * h.

template <int Kin, int F, bool RELU_IN>
__global__ void gemm_wmma_f32(const float* __restrict__ A,
                              const float* __restrict__ W,
                              const float* __restrict__ dinv,
                              const float* __restrict__ bias,
                              float* __restrict__ H,
                              float* __restrict__ OutInit, int nRows) {
  const int gwave = (blockIdx.x * blockDim.x + threadIdx.x) >> 5;
  const int lane = threadIdx.x & 31;
  const int nRowTiles = (nRows + 15) >> 4;
  if (gwave >= nRowTiles) return;  // uniform per wave: EXEC stays all-ones

  const int row0 = gwave << 4;
  const int m = lane & 15;                 // A row within tile / B-C column
  const int khalf = (lane >> 4) << 1;      // lanes 0-15 -> k+{0,1}; 16-31 -> k+{2,3}
  const int arow = row0 + m;
  const int arowc = (arow < nRows) ? arow : (nRows - 1);  // clamp (tail-safe)

  constexpr int KT = (Kin + 3) / 4;        // K padded to multiple of 4 via masks
  constexpr int NT = (F + 15) / 16;

#pragma unroll
  for (int nt = 0; nt < NT; ++nt) {
    const int n0 = nt * 16;
    const int nn = n0 + m;
    const float bmask = (nn < F) ? 1.0f : 0.0f;
    const int nns = (nn < F) ? nn : (F - 1);

    v8f c = {};
#pragma unroll
    for (int kt = 0; kt < KT; ++kt) {
      const int k0 = kt * 4 + khalf;
      const int k1 = k0 + 1;
      const float m0 = (k0 < Kin) ? 1.0f : 0.0f;
      const float m1 = (k1 < Kin) ? 1.0f : 0.0f;
      const int k0s = (k0 < Kin) ? k0 : 0;
      const int k1s = (k1 < Kin) ? k1 : 0;

      // A fragment (16x4 f32: 2 VGPRs per lane)
      float a0 = A[arowc * Kin + k0s];
      float a1 = A[arowc * Kin + k1s];
      if (RELU_IN) { a0 = fmaxf(a0, 0.0f); a1 = fmaxf(a1, 0.0f); }
      v2f a;
      a.x = a0 * m0;
      a.y = a1 * m1;

      // B fragment (4x16 f32: 2 VGPRs per lane, row striped across lanes)
      v2f b;
      b.x = W[k0s * F + nns] * (m0 * bmask);
      b.y = W[k1s * F + nns] * (m1 * bmask);

      // D = A*B + C   (8 args: neg_a, A, neg_b, B, c_mod, C, reuse_a, reuse_b)
      c = __builtin_amdgcn_wmma_f32_16x16x4_f32(false, a, false, b, (short)0, c,
                                                false, false);
    }

    // C/D layout: v8f element r -> row (r + (lane<16?0:8)), col = m.
    // Fast path: whole 16-row tile in range -> branch-free unrolled stores
    // (avoids per-row v_cmpx/exec churn; this is always taken for N=100000).
    if (nn < F) {
      const int rbase = row0 + ((lane < 16) ? 0 : 8);
      const float bf = bias[nn];
      if (row0 + 16 <= nRows) {
#pragma unroll
        for (int r = 0; r < 8; ++r) {
          const int row = rbase + r;
          const float h = c[r];
          const float w = dinv[row];
          H[row * F + nn] = h;
          OutInit[row * F + nn] = bf + w * w * h;
        }
      } else {
#pragma unroll
        for (int r = 0; r < 8; ++r) {
          const int row = rbase + r;
          if (row < nRows) {
            const float h = c[r];
            const float w = dinv[row];
            H[row * F + nn] = h;
            OutInit[row * F + nn] = bf + w * w * h;
          }
        }
      }
    }
  }
}

// ---------------- edge scatter: out[dst] += dinv[src]*dinv[dst] * h[src] --------

template <int F>
__global__ void edge_scatter(const int* __restrict__ src, const int* __restrict__ dst,
                             const float* __restrict__ dinv,
                             const float* __restrict__ h, float* __restrict__ out,
                             int e) {
  int t = blockIdx.x * TPB + threadIdx.x;
  if (t >= e * F) return;
  int ed = t / F;       // consecutive threads share an edge -> coalesced rows
  int f = t - ed * F;
  int s = src[ed];
  int d = dst[ed];
  float w = dinv[s] * dinv[d];
  unsafeAtomicAdd(&out[d * F + f], w * h[s * F + f]);
}

// ---------------- host side ----------------

extern "C" void kernel_launch(void* const* d_in, const int* in_sizes, int n_in,
                              void* d_out, int out_size, void* d_ws, size_t ws_size,
                              hipStream_t stream) {
  const float* x  = (const float*)d_in[0];
  const int*  ei  = (const int*)d_in[1];
  const float* W1 = (const float*)d_in[2];
  const float* b1 = (const float*)d_in[3];
  const float* W2 = (const float*)d_in[4];
  const float* b2 = (const float*)d_in[5];
  const float* W3 = (const float*)d_in[6];
  const float* b3 = (const float*)d_in[7];

  const int N = in_sizes[0] / 3;   // 100000
  const int E = in_sizes[1] / 2;   // 3200000
  const int* src = ei;
  const int* dst = ei + E;

  float* dinv = (float*)d_ws;                  // N
  float* bufA = dinv + N;                      // N x 32 (transform output h)
  float* bufB = bufA + (size_t)N * 32;         // N x 32 (aggregated layer output)
  float* out  = (float*)d_out;                 // N x 2

  auto cdiv = [](int a, int b) { return (a + b - 1) / b; };

  // symmetric-norm coefficients (computed fresh every call; ws fully rewritten)
  init_deg<<<cdiv(N, TPB), TPB, 0, stream>>>(dinv, N);
  accum_deg<<<cdiv(E, TPB), TPB, 0, stream>>>(dst, dinv, E);
  finish_dinv<<<cdiv(N, TPB), TPB, 0, stream>>>(dinv, N);

  const int rowTiles = cdiv(N, 16);
  const int gemmBlocks = cdiv(rowTiles, TPB / 32);  // 8 waves per 256-thread block

  // layer 1: 3 -> 16 (GEMM also writes bufB = b1 + dinv^2 * h, the self-loop term)
  gemm_wmma_f32<3, 16, false><<<gemmBlocks, TPB, 0, stream>>>(x, W1, dinv, b1,
                                                              bufA, bufB, N);
  edge_scatter<16><<<cdiv(E * 16, TPB), TPB, 0, stream>>>(src, dst, dinv, bufA, bufB, E);

  // layer 2: 16 -> 32 (ReLU fused into A-fragment load)
  gemm_wmma_f32<16, 32, true><<<gemmBlocks, TPB, 0, stream>>>(bufB, W2, dinv, b2,
                                                              bufA, bufB, N);
  edge_scatter<32><<<cdiv(E * 32, TPB), TPB, 0, stream>>>(src, dst, dinv, bufA, bufB, E);

  // layer 3: 32 -> 2 (ReLU fused), aggregate straight into d_out
  gemm_wmma_f32<32, 2, true><<<gemmBlocks, TPB, 0, stream>>>(bufB, W3, dinv, b3,
                                                             bufA, out, N);
  edge_scatter<2><<<cdiv(E * 2, TPB), TPB, 0, stream>>>(src, dst, dinv, bufA, out, E);
}